// KDAUpdate_74079595921664
// MI455X (gfx1250) — compile-verified
//
#include <hip/hip_runtime.h>

// KDA state update, MI455X (gfx1250), wave32.
// S_new = Diag(a)S - b k (k^T Diag(a) S) + b k v^T ;  o = S_new^T q
//       = Diag(a)S + k * bm^T,  bm = b*(v - kT_S),  o = qS + (k.q)*bm
// One 128-thread workgroup (4 wave32) per (b,h). Dual matvec (kT_S, qS)
// via V_WMMA_F32_16X16X4_F32. Elementwise update pass uses b128 streams
// with non-temporal hints (output is write-once, re-read of S is last-use).

#define KDA_B  64
#define KDA_H  32
#define KDA_DK 128
#define KDA_DV 128

typedef __attribute__((ext_vector_type(2))) float v2f;
typedef __attribute__((ext_vector_type(4))) float f4;
typedef __attribute__((ext_vector_type(8))) float v8f;

__global__ __launch_bounds__(128) void KDAUpdate_74079595921664_kernel(
    const float* __restrict__ S, const float* __restrict__ k,
    const float* __restrict__ v, const float* __restrict__ q,
    const float* __restrict__ alpha, const float* __restrict__ beta,
    float* __restrict__ Sout, float* __restrict__ o)
{
    const int bh = blockIdx.x;          // 0..2047
    const int t  = threadIdx.x;         // 0..127
    const float* __restrict__ Sbh = S    + (size_t)bh * KDA_DK * KDA_DV;
    float*       __restrict__ So  = Sout + (size_t)bh * KDA_DK * KDA_DV;

    // abase: [0..127]=alpha*k, [128..255]=alpha*q, [256..383]=zeros
    __shared__ float abase[3 * KDA_DK];
    __shared__ float k_s[KDA_DK], al_s[KDA_DK], v_s[KDA_DV];
    __shared__ float red[KDA_DK];
    __shared__ float kt_s[KDA_DV], qs_s[KDA_DV];
    __shared__ __align__(16) float bm_s[KDA_DV];

    // ---- stage small vectors, build WMMA A-operand table ----
    const float kv  = k[(size_t)bh * KDA_DK + t];
    const float qv  = q[(size_t)bh * KDA_DK + t];
    const float av  = alpha[(size_t)bh * KDA_DK + t];
    const float vv  = v[(size_t)bh * KDA_DV + t];
    const float bet = beta[bh];
    k_s[t]  = kv;
    al_s[t] = av;
    v_s[t]  = vv;
    abase[t]              = kv * av;   // k' = alpha .* k
    abase[KDA_DK + t]     = qv * av;   // q' = alpha .* q
    abase[2 * KDA_DK + t] = 0.0f;      // zero row for A rows 2..15
    red[t] = kv * qv;                  // for kq = dot(k, q)
    __syncthreads();

    // tree-reduce kq over 128 threads
    for (int s = 64; s > 0; s >>= 1) {
        if (t < s) red[t] += red[t + s];
        __syncthreads();
    }
    const float kq = red[0];

    // ---- phase 1: dual matvec via V_WMMA_F32_16X16X4_F32 ----
    // C[m][n] = sum_d A[m][d] * S[d, c0+n]; row0 = kT_S, row1 = qS.
    // A layout (16x4 f32): lane(0..15)=M, VGPR{0,1} x lane-half = K{0,1,2,3}.
    // B layout (4x16 f32, mirror of A): lane(0..15)=N, VGPR{0,1} x half = K.
    // C/D layout: vgpr r, lanes 0..15 -> row M=r, N=lane.
    const int wave = t >> 5;          // 0..3, each wave owns 32 columns
    const int lane = t & 31;
    const int row  = lane & 15;
    const int half = lane >> 4;       // 0 -> K{0,1}, 1 -> K{2,3}
    const int asel = (row < 2) ? row : 2;
    const float* __restrict__ Abase = &abase[asel * KDA_DK];

    v8f acc0 = {};   // columns [wave*32,      wave*32 + 16)
    v8f acc1 = {};   // columns [wave*32 + 16, wave*32 + 32)
    const int c0 = wave * 32;

    for (int d0 = 0; d0 < KDA_DK; d0 += 4) {
        v2f a;
        a.x = Abase[d0 + 2 * half];
        a.y = Abase[d0 + 2 * half + 1];

        const float* __restrict__ bp0 =
            Sbh + (size_t)(d0 + 2 * half) * KDA_DV + c0 + row;
        v2f b0, b1;
        b0.x = bp0[0];           // row d0+2*half,   col c0+row
        b0.y = bp0[KDA_DV];      // row d0+2*half+1, col c0+row
        b1.x = bp0[16];          // second 16-column tile
        b1.y = bp0[KDA_DV + 16];

        acc0 = __builtin_amdgcn_wmma_f32_16x16x4_f32(
            false, a, false, b0, (short)0, acc0, false, false);
        acc1 = __builtin_amdgcn_wmma_f32_16x16x4_f32(
            false, a, false, b1, (short)0, acc1, false, false);
    }

    // rows 0/1 of C live in vgpr 0/1, lanes 0..15
    if (lane < 16) {
        kt_s[c0 + lane]      = acc0[0];
        qs_s[c0 + lane]      = acc0[1];
        kt_s[c0 + 16 + lane] = acc1[0];
        qs_s[c0 + 16 + lane] = acc1[1];
    }
    __syncthreads();

    // ---- outputs: o, bm table ----
    const float bm = bet * (v_s[t] - kt_s[t]);      // bm[t]
    bm_s[t] = bm;
    o[(size_t)bh * KDA_DV + t] = qs_s[t] + kq * bm;
    __syncthreads();

    // ---- phase 2: S_new = Diag(a)S + k*bm^T, b128 streaming ----
    // Thread owns columns c4..c4+3; wave streams one full 512B row/iter.
    // Row for iter i: d = 4*i + wave (uniform within wave).
    const int c4 = (lane) * 4;
    const f4 bm4 = *(const f4*)&bm_s[c4];
    #pragma unroll 4
    for (int i = 0; i < 32; ++i) {
        const int d = 4 * i + wave;
        const float ad = al_s[d];
        const float kd = k_s[d];
        const f4 s4 = __builtin_nontemporal_load(
            (const f4*)(Sbh + (size_t)d * KDA_DV + c4));   // last use of S
        f4 r = ad * s4 + kd * bm4;
        __builtin_nontemporal_store(
            r, (f4*)(So + (size_t)d * KDA_DV + c4));       // write-once stream
    }
}

extern "C" void kernel_launch(void* const* d_in, const int* in_sizes, int n_in,
                              void* d_out, int out_size, void* d_ws, size_t ws_size,
                              hipStream_t stream) {
    const float* S     = (const float*)d_in[0];
    const float* k     = (const float*)d_in[1];
    const float* v     = (const float*)d_in[2];
    const float* q     = (const float*)d_in[3];
    const float* alpha = (const float*)d_in[4];
    const float* beta  = (const float*)d_in[5];

    float* Sout = (float*)d_out;
    float* o    = Sout + (size_t)KDA_B * KDA_H * KDA_DK * KDA_DV;

    dim3 grid(KDA_B * KDA_H);
    dim3 block(128);
    KDAUpdate_74079595921664_kernel<<<grid, block, 0, stream>>>(
        S, k, v, q, alpha, beta, Sout, o);
}